// constraints_enforcement_78847009620602
// MI455X (gfx1250) — compile-verified
//
#include <hip/hip_runtime.h>
#include <stdint.h>

// ---------------------------------------------------------------------------
// Batched Rodrigues rotation (R rotates unit(vec1) onto unit(vec2)).
// Memory-bound streaming kernel: stage inputs/outputs through LDS with the
// CDNA5 async global<->LDS data movers so every global transaction is B128.
// ---------------------------------------------------------------------------

#define EPW  128              // elements per wave (wave-private LDS tile)
#define WPB  8                // waves per block (256 threads, wave32)
#define EPB  (EPW * WPB)      // 1024 elements per block

// Low 32 bits of a generic pointer into LDS == workgroup-relative LDS offset.
__device__ __forceinline__ uint32_t lds_offset(const void* p) {
  return (uint32_t)(uintptr_t)p;
}

// CDNA5 async copy: global memory -> LDS, 128 bits per lane, ASYNCcnt-tracked.
__device__ __forceinline__ void async_load_b128(uint32_t lds, const void* g) {
  asm volatile("global_load_async_to_lds_b128 %0, %1, off"
               :: "v"(lds), "v"(g) : "memory");
}
// CDNA5 async copy: LDS -> global memory, 128 bits per lane.
__device__ __forceinline__ void async_store_b128(const void* g, uint32_t lds) {
  asm volatile("global_store_async_from_lds_b128 %0, %1, off"
               :: "v"(g), "v"(lds) : "memory");
}
__device__ __forceinline__ void wait_async0() {
  asm volatile("s_wait_asynccnt 0x0" ::: "memory");
}
__device__ __forceinline__ void wait_ds0() {
  asm volatile("s_wait_dscnt 0x0" ::: "memory");
}

// One Rodrigues rotation, replicating the reference's branchless select chain.
__device__ __forceinline__ void rodrigues_elem(float ax, float ay, float az,
                                               float bx, float by, float bz,
                                               float* __restrict__ R) {
  // normalize both vectors
  const float i1 = 1.0f / sqrtf(ax * ax + ay * ay + az * az);
  const float i2 = 1.0f / sqrtf(bx * bx + by * by + bz * bz);
  ax *= i1; ay *= i1; az *= i1;
  bx *= i2; by *= i2; bz *= i2;

  // v = a x b, c = a.b, s = |v|
  const float vx = ay * bz - az * by;
  const float vy = az * bx - ax * bz;
  const float vz = ax * by - ay * bx;
  const float c  = ax * bx + ay * by + az * bz;
  const float s2 = vx * vx + vy * vy + vz * vz;
  const float s  = sqrtf(s2);

  const bool degen = (s < 1e-30f);
  const float t = (1.0f - c) / (degen ? 1.0f : s2);  // s2_safe

  // R = I + K + t*(v v^T - s2*I)   (K@K == v v^T - |v|^2 I)
  float Rm[9];
  Rm[0] = 1.0f + t * (vx * vx - s2);
  Rm[1] = -vz  + t * (vx * vy);
  Rm[2] =  vy  + t * (vx * vz);
  Rm[3] =  vz  + t * (vy * vx);
  Rm[4] = 1.0f + t * (vy * vy - s2);
  Rm[5] = -vx  + t * (vy * vz);
  Rm[6] = -vy  + t * (vz * vx);
  Rm[7] =  vx  + t * (vz * vy);
  Rm[8] = 1.0f + t * (vz * vz - s2);

  // 180-degree branch: axis = allclose(a, e1) ? e2 : e1 ; perp = a x axis
  const bool close = (fabsf(ax - 1.0f) <= (1e-8f + 1e-5f)) &&
                     (fabsf(ay) <= 1e-8f) && (fabsf(az) <= 1e-8f);
  const float exv = close ? 0.0f : 1.0f;
  const float eyv = close ? 1.0f : 0.0f;
  float px = -az * eyv;            // ay*ez - az*ey  (ez == 0)
  float py =  az * exv;            // az*ex - ax*ez
  float pz =  ax * eyv - ay * exv; // ax*ey - ay*ex
  const float pn = sqrtf(px * px + py * py + pz * pz);
  const float dn = (pn == 0.0f) ? 1.0f : pn;
  px /= dn; py /= dn; pz /= dn;
  const float q2 = px * px + py * py + pz * pz;
  float R180[9];
  R180[0] = 1.0f + 2.0f * (px * px - q2);
  R180[1] = 2.0f * px * py;
  R180[2] = 2.0f * px * pz;
  R180[3] = 2.0f * py * px;
  R180[4] = 1.0f + 2.0f * (py * py - q2);
  R180[5] = 2.0f * py * pz;
  R180[6] = 2.0f * pz * px;
  R180[7] = 2.0f * pz * py;
  R180[8] = 1.0f + 2.0f * (pz * pz - q2);

  const bool sel_eye = degen && (c > 0.0f);
  const bool sel_180 = degen && (c < 0.0f);
  const float I9[9] = {1.0f, 0.0f, 0.0f, 0.0f, 1.0f, 0.0f, 0.0f, 0.0f, 1.0f};
#pragma unroll
  for (int k = 0; k < 9; ++k)
    R[k] = sel_eye ? I9[k] : (sel_180 ? R180[k] : Rm[k]);
}

extern "C" __global__ void __launch_bounds__(256)
rodrigues_async_kernel(const float* __restrict__ v1,
                       const float* __restrict__ v2,
                       float* __restrict__ out, long long total) {
  // Wave-private LDS tiles (no cross-wave sharing => no workgroup barriers).
  __shared__ __align__(16) float s_in1[WPB * EPW * 3];  // 12 KB
  __shared__ __align__(16) float s_in2[WPB * EPW * 3];  // 12 KB
  __shared__ __align__(16) float s_out[WPB * EPW * 9];  // 36 KB

  const int wid  = (int)(threadIdx.x >> 5);   // wave32
  const int lane = (int)(threadIdx.x & 31);
  const long long we0 = (long long)blockIdx.x * EPB + (long long)wid * EPW;

  if (we0 + EPW <= total) {
    // --- stage inputs: 1536 B per array per wave = 3 x b128 per lane -------
    const char* g1 = (const char*)(v1 + we0 * 3);
    const char* g2 = (const char*)(v2 + we0 * 3);
    const uint32_t l1 = lds_offset(&s_in1[wid * EPW * 3]);
    const uint32_t l2 = lds_offset(&s_in2[wid * EPW * 3]);
#pragma unroll
    for (int r = 0; r < 3; ++r) {
      const int o = r * 512 + lane * 16;      // 32 lanes x 16 B = 512 B/round
      async_load_b128(l1 + o, g1 + o);
      async_load_b128(l2 + o, g2 + o);
    }
    wait_async0();  // inputs resident in this wave's LDS tile

    // --- compute 4 elements per lane from LDS ------------------------------
    const float* in1 = &s_in1[wid * EPW * 3];
    const float* in2 = &s_in2[wid * EPW * 3];
    float* so = &s_out[wid * EPW * 9];
#pragma unroll
    for (int j = 0; j < 4; ++j) {
      const int e = lane + 32 * j;
      float R[9];
      rodrigues_elem(in1[3 * e + 0], in1[3 * e + 1], in1[3 * e + 2],
                     in2[3 * e + 0], in2[3 * e + 1], in2[3 * e + 2], R);
#pragma unroll
      for (int k = 0; k < 9; ++k) so[9 * e + k] = R[k];
    }

    // ds stores retire on DScnt; async engine is a separate pipe -> fence it.
    wait_ds0();

    // --- drain output tile: 4608 B per wave = 9 x b128 per lane ------------
    const char* go = (const char*)(out + we0 * 9);
    const uint32_t lo = lds_offset(so);
#pragma unroll
    for (int r = 0; r < 9; ++r) {
      const int o = r * 512 + lane * 16;
      async_store_b128(go + o, lo + o);
    }
    // s_endpgm performs an implicit wait-idle, covering outstanding ASYNCcnt.
  } else {
    // Scalar tail (dead for the reference sizes: total % 1024 == 0).
    const long long end = (we0 + EPW < total) ? (we0 + EPW) : total;
    for (long long e = we0 + lane; e < end; e += 32) {
      float R[9];
      rodrigues_elem(v1[3 * e + 0], v1[3 * e + 1], v1[3 * e + 2],
                     v2[3 * e + 0], v2[3 * e + 1], v2[3 * e + 2], R);
      for (int k = 0; k < 9; ++k) out[9 * e + k] = R[k];
    }
  }
}

extern "C" void kernel_launch(void* const* d_in, const int* in_sizes, int n_in,
                              void* d_out, int out_size, void* d_ws, size_t ws_size,
                              hipStream_t stream) {
  const float* v1 = (const float*)d_in[0];
  const float* v2 = (const float*)d_in[1];
  float* out = (float*)d_out;
  const long long total = (long long)in_sizes[0] / 3;   // B*N elements
  const int blocks = (int)((total + EPB - 1) / EPB);
  rodrigues_async_kernel<<<blocks, 256, 0, stream>>>(v1, v2, out, total);
  (void)n_in; (void)out_size; (void)d_ws; (void)ws_size;
}